// Hed_19885698580968
// MI455X (gfx1250) — compile-verified
//
#include <hip/hip_runtime.h>

typedef float v2f __attribute__((ext_vector_type(2)));
typedef float v8f __attribute__((ext_vector_type(8)));

#define B_   8
#define NN   4096            // N*N = 64*64
#define ROWS (B_ * NN)       // 32768 edge rows per tensor

// ------------------------------------------------------------------
// Kernel 1: Y[b,e,i] = sum_j W_edge_sub[i,j] * am1[b,e,j]
// ------------------------------------------------------------------
__global__ void y_kernel(const float* __restrict__ am1,
                         const float* __restrict__ Wsub,
                         float* __restrict__ Y) {
    int gid = blockIdx.x * blockDim.x + threadIdx.x;
    if (gid >= ROWS) return;
    const float* a = am1 + (long)gid * 16;
    float v[16];
#pragma unroll
    for (int j = 0; j < 16; ++j) v[j] = a[j];
#pragma unroll
    for (int i = 0; i < 16; ++i) {
        float t = 0.f;
#pragma unroll
        for (int j = 0; j < 16; ++j) t += Wsub[i * 16 + j] * v[j];
        Y[(long)gid * 16 + i] = t;
    }
}

// ------------------------------------------------------------------
// Kernel 2: per-edge quadratic forms (the min seeds + d contributions)
//   gid < ROWS      : e_del1 over am1 rows with W_edge_del
//   ROWS..2*ROWS-1  : e_ins2 over am2 rows with W_edge_ins
// ------------------------------------------------------------------
__global__ void init_edge_kernel(const float* __restrict__ am1,
                                 const float* __restrict__ am2,
                                 const float* __restrict__ Wdel,
                                 const float* __restrict__ Wins,
                                 float* __restrict__ eDel,
                                 float* __restrict__ eIns) {
    int gid = blockIdx.x * blockDim.x + threadIdx.x;
    if (gid >= 2 * ROWS) return;
    bool second = gid >= ROWS;
    int r = second ? (gid - ROWS) : gid;
    const float* a = (second ? am2 : am1) + (long)r * 16;
    const float* W = second ? Wins : Wdel;
    float v[16];
#pragma unroll
    for (int j = 0; j < 16; ++j) v[j] = a[j];
    float q = 0.f;
#pragma unroll
    for (int i = 0; i < 16; ++i) {
        float t = 0.f;
#pragma unroll
        for (int j = 0; j < 16; ++j) t += W[i * 16 + j] * v[j];
        q += v[i] * t;
    }
    if (second) eIns[r] = q; else eDel[r] = q;
}

// ------------------------------------------------------------------
// Kernel 3: node insertion cost v2^T W_node_ins v2 per (b,n)
// ------------------------------------------------------------------
__global__ void node2_kernel(const float* __restrict__ v2,
                             const float* __restrict__ Wni,
                             float* __restrict__ node2) {
    int gid = blockIdx.x * blockDim.x + threadIdx.x;
    if (gid >= B_ * 64) return;
    const float* v = v2 + (long)gid * 64;
    float acc = 0.f;
    for (int i = 0; i < 64; ++i) {
        float t = 0.f;
#pragma unroll 8
        for (int j = 0; j < 64; ++j) t += Wni[i * 64 + j] * v[j];
        acc += v[i] * t;
    }
    node2[gid] = acc;
}

// ------------------------------------------------------------------
// Fragment loader: 16-row x 16-elem f32 tile, ISA 16x4 f32 striping.
// Lane l<16 holds row l, K = 4s,4s+1 ; lane>=16 holds row l-16,
// K = 4s+2,4s+3.  Same per-lane pattern serves A- and B-operands.
// rowbase = tile_base + l16*16 + (hi?2:0)
// ------------------------------------------------------------------
__device__ __forceinline__ void load_frags(const float* __restrict__ rowbase,
                                           v2f f[4]) {
#pragma unroll
    for (int s = 0; s < 4; ++s)
        f[s] = *(const v2f*)(rowbase + 4 * s);   // global_load_b64, imm offsets
}

// ------------------------------------------------------------------
// Kernel 4 (run twice with swapped operands):
//   C[m][n] = sum_i Arows[t*16+m][i] * Brows[ft*16+n][i]   (= 2*csub)
//   outmin[row] = min( 0.5 * min_n_over_all_tiles C, seed[row] )
// Pass 1: A=Y,   B=am2 -> rowmin (min over f), seed = e_del1
// Pass 2: A=am2, B=Y   -> colmin (min over e), seed = e_ins2
// One wave per 16-row A-tile; no LDS, no barriers, no atomics.
// ------------------------------------------------------------------
__global__ __launch_bounds__(256) void pass_min_kernel(
        const float* __restrict__ Arows, const float* __restrict__ Brows,
        const float* __restrict__ seed, float* __restrict__ outmin) {
    const int tid  = threadIdx.x;
    const int lane = tid & 31;
    const int wv   = tid >> 5;
    const bool hi  = lane >= 16;
    const int l16  = lane & 15;
    const int b    = blockIdx.y;
    const int t    = blockIdx.x * 8 + wv;        // A-tile index 0..255
    const long base = (long)b * NN;
    const int  koff = hi ? 2 : 0;

    // A fragments: loaded once, reused for all 256 B-tiles
    v2f af[4];
    load_frags(Arows + (base + t * 16 + l16) * 16 + koff, af);

    float rm[8];
#pragma unroll
    for (int g = 0; g < 8; ++g) rm[g] = __builtin_inff();

    const float* brow = Brows + (base + l16) * 16 + koff;

    // software-pipelined: prefetch next tile's fragments before WMMAs
    v2f cb[4], nb[4];
    load_frags(brow, cb);
    for (int ft = 0; ft < 256; ++ft) {
        if (ft + 1 < 256) load_frags(brow + (long)(ft + 1) * 256, nb);
        v8f c = {};
#pragma unroll
        for (int s = 0; s < 4; ++s)
            c = __builtin_amdgcn_wmma_f32_16x16x4_f32(
                    false, af[s], false, cb[s], (short)0, c, false, false);
#pragma unroll
        for (int g = 0; g < 8; ++g) rm[g] = fminf(rm[g], c[g]);
#pragma unroll
        for (int s = 0; s < 4; ++s) cb[s] = nb[s];
    }

    // reduce min across the 16 lanes of each half; fold seed; plain store
#pragma unroll
    for (int g = 0; g < 8; ++g) {
        float v = rm[g];
        v = fminf(v, __shfl_xor(v, 8, 32));
        v = fminf(v, __shfl_xor(v, 4, 32));
        v = fminf(v, __shfl_xor(v, 2, 32));
        v = fminf(v, __shfl_xor(v, 1, 32));
        if (l16 == 0) {                          // lane 0 -> M=g, lane 16 -> M=g+8
            long r = base + t * 16 + g + (hi ? 8 : 0);
            outmin[r] = fminf(0.5f * v, seed[r]);
        }
    }
}

// ------------------------------------------------------------------
// Kernel 5: s1[b,i] = sum_j rowmin ; s2[b,j] = sum_j colmin
// ------------------------------------------------------------------
__global__ void sum_rc_kernel(const float* __restrict__ rowmin,
                              const float* __restrict__ colmin,
                              float* __restrict__ s1, float* __restrict__ s2) {
    int gid = blockIdx.x * blockDim.x + threadIdx.x;
    if (gid >= 2 * B_ * 64) return;
    bool second = gid >= B_ * 64;
    int g = second ? gid - B_ * 64 : gid;
    int b = g >> 6, i = g & 63;
    const float* src = second ? colmin : rowmin;
    float s = 0.f;
#pragma unroll 8
    for (int j = 0; j < 64; ++j) s += src[(long)b * NN + i * 64 + j];
    if (second) s2[g] = s; else s1[g] = s;
}

// ------------------------------------------------------------------
// Kernel 6: ce[b,i,j] = s1[b,i] + s2[b,j]  -> out[1..]
// ------------------------------------------------------------------
__global__ void ce_kernel(const float* __restrict__ s1,
                          const float* __restrict__ s2,
                          float* __restrict__ out) {
    int idx = blockIdx.x * blockDim.x + threadIdx.x;
    if (idx >= B_ * 64 * 64) return;
    int b = idx >> 12, i = (idx >> 6) & 63, j = idx & 63;
    out[1 + idx] = s1[b * 64 + i] + s2[b * 64 + j];
}

// ------------------------------------------------------------------
// Kernel 7: deterministic scalar d (single block, fixed-order tree)
// d = sum v1^2 + sum node2 + 0.5*sum eDel + 0.5*sum eIns
// ------------------------------------------------------------------
__global__ void d_kernel(const float* __restrict__ v1,
                         const float* __restrict__ eDel,
                         const float* __restrict__ eIns,
                         const float* __restrict__ node2,
                         float* __restrict__ out) {
    __shared__ float red[256];
    int tid = threadIdx.x;
    float acc = 0.f;
    for (int k = tid; k < ROWS; k += 256) acc += v1[k] * v1[k];   // |v1| = 32768
    for (int k = tid; k < B_ * 64; k += 256) acc += node2[k];
    for (int k = tid; k < ROWS; k += 256) acc += 0.5f * eDel[k];
    for (int k = tid; k < ROWS; k += 256) acc += 0.5f * eIns[k];
    red[tid] = acc;
    __syncthreads();
    for (int s = 128; s > 0; s >>= 1) {
        if (tid < s) red[tid] += red[tid + s];
        __syncthreads();
    }
    if (tid == 0) out[0] = red[0];
}

// ------------------------------------------------------------------
extern "C" void kernel_launch(void* const* d_in, const int* in_sizes, int n_in,
                              void* d_out, int out_size, void* d_ws, size_t ws_size,
                              hipStream_t stream) {
    const float* v1   = (const float*)d_in[0];
    const float* am1  = (const float*)d_in[1];
    const float* v2   = (const float*)d_in[3];
    const float* am2  = (const float*)d_in[4];
    const float* Wni  = (const float*)d_in[6];   // W_node_ins
    const float* Wei  = (const float*)d_in[9];   // W_edge_ins
    const float* Wed  = (const float*)d_in[10];  // W_edge_del
    const float* Wes  = (const float*)d_in[11];  // W_edge_sub
    float* out = (float*)d_out;

    float* ws = (float*)d_ws;
    float* Y      = ws;                       // 524288  (2 MB)
    float* eDel   = Y + (long)ROWS * 16;      // 32768
    float* eIns   = eDel + ROWS;              // 32768
    float* rowmin = eIns + ROWS;              // 32768
    float* colmin = rowmin + ROWS;            // 32768
    float* node2  = colmin + ROWS;            // 512
    float* s1     = node2 + B_ * 64;          // 512
    float* s2     = s1 + B_ * 64;             // 512

    y_kernel<<<(ROWS + 255) / 256, 256, 0, stream>>>(am1, Wes, Y);
    init_edge_kernel<<<(2 * ROWS + 255) / 256, 256, 0, stream>>>(
        am1, am2, Wed, Wei, eDel, eIns);
    node2_kernel<<<2, 256, 0, stream>>>(v2, Wni, node2);

    // pass 1: rowmin over f (A = Y, B = am2), seeded with e_del1
    pass_min_kernel<<<dim3(32, B_), 256, 0, stream>>>(Y, am2, eDel, rowmin);
    // pass 2: colmin over e (A = am2, B = Y), seeded with e_ins2
    pass_min_kernel<<<dim3(32, B_), 256, 0, stream>>>(am2, Y, eIns, colmin);

    sum_rc_kernel<<<4, 256, 0, stream>>>(rowmin, colmin, s1, s2);
    ce_kernel<<<(B_ * 64 * 64 + 255) / 256, 256, 0, stream>>>(s1, s2, out);
    d_kernel<<<1, 256, 0, stream>>>(v1, eDel, eIns, node2, out);
}